// SE_GATNoEdgeAttrs_44641890075098
// MI455X (gfx1250) — compile-verified
//
#include <hip/hip_runtime.h>
#include <math.h>

typedef __attribute__((ext_vector_type(2))) float v2f;
typedef __attribute__((ext_vector_type(8))) float v8f;

#define NEG_SLOPE 0.2f

// ---------------------------------------------------------------- utilities
__global__ void fill_kernel(float* __restrict__ p, float v, long n) {
  long i = (long)blockIdx.x * blockDim.x + threadIdx.x;
  long stride = (long)gridDim.x * blockDim.x;
  for (; i < n; i += stride) p[i] = v;
}

__device__ inline void atomicMaxFloat(float* addr, float val) {
  // monotone bit-ordering trick: ints order nonneg floats, uints reverse-order neg floats
  if (val >= 0.0f)
    atomicMax((int*)addr, __float_as_int(val));
  else
    atomicMin((unsigned int*)addr, __float_as_uint(val));
}

// ------------------------------------------------- fp32 WMMA GEMM, row-major
// C[M,Ncols] = A[M,K] * B[K,Ncols].
// One wave computes a 16x64 output strip (4 WMMA accumulators) so the A
// fragment is loaded once per K-step and reused across 4 WMMAs.
// grid = (ceil(M/16), Ncols/64), block = 32 (1 wave).
__global__ void gemm_wmma_f32_kernel(const float* __restrict__ A,
                                     const float* __restrict__ B,
                                     float* __restrict__ C,
                                     int M, int K, int Ncols) {
  const int lane = threadIdx.x;
  const int l15  = lane & 15;
  const int lhi  = lane >> 4;        // 0 or 1
  const int koff = lhi * 2;          // K sub-offset held by this half-wave

  int arow = blockIdx.x * 16 + l15;
  if (arow >= M) arow = M - 1;       // clamp loads; stores are guarded
  const float* Ap = A + (long)arow * K;
  const int nbase = blockIdx.y * 64;
  const float* Bp = B + nbase + l15;

  v8f acc0 = {}, acc1 = {}, acc2 = {}, acc3 = {};
  for (int k0 = 0; k0 < K; k0 += 4) {
    v2f a;
    a.x = Ap[k0 + koff];                       // A[m][k0+2g]
    a.y = Ap[k0 + koff + 1];                   // A[m][k0+2g+1]
    const float* Bk0 = Bp + (long)(k0 + koff) * Ncols;
    const float* Bk1 = Bk0 + Ncols;
    v2f b0, b1, b2, b3;
    b0.x = Bk0[0];  b0.y = Bk1[0];             // n-tile 0
    b1.x = Bk0[16]; b1.y = Bk1[16];            // n-tile 1
    b2.x = Bk0[32]; b2.y = Bk1[32];            // n-tile 2
    b3.x = Bk0[48]; b3.y = Bk1[48];            // n-tile 3
    acc0 = __builtin_amdgcn_wmma_f32_16x16x4_f32(false, a, false, b0, (short)0, acc0, false, false);
    acc1 = __builtin_amdgcn_wmma_f32_16x16x4_f32(false, a, false, b1, (short)0, acc1, false, false);
    acc2 = __builtin_amdgcn_wmma_f32_16x16x4_f32(false, a, false, b2, (short)0, acc2, false, false);
    acc3 = __builtin_amdgcn_wmma_f32_16x16x4_f32(false, a, false, b3, (short)0, acc3, false, false);
  }

  const int mbase = blockIdx.x * 16 + lhi * 8;
#pragma unroll
  for (int v = 0; v < 8; ++v) {
    int mrow = mbase + v;
    if (mrow < M) {
      float* Cp = C + (long)mrow * Ncols + nbase + l15;
      Cp[0]  = acc0[v];
      Cp[16] = acc1[v];
      Cp[32] = acc2[v];
      Cp[48] = acc3[v];
    }
  }
}

// --------------------------------------------- per-node attention logits a_s, a_d
// one thread per (node, head); h layout [N, 4*C], a_* layout [4, C]
__global__ void alpha_kernel(const float* __restrict__ h,
                             const float* __restrict__ a_src,
                             const float* __restrict__ a_dst,
                             float* __restrict__ as, float* __restrict__ ad,
                             int N, int C) {
  int i = blockIdx.x * blockDim.x + threadIdx.x;   // i = n*4 + head
  if (i >= N * 4) return;
  int head = i & 3;
  int n = i >> 2;
  const float* hp = h + (long)n * 4 * C + head * C;
  const float* s  = a_src + head * C;
  const float* d  = a_dst + head * C;
  float accs = 0.0f, accd = 0.0f;
  for (int c = 0; c < C; ++c) { float v = hp[c]; accs += v * s[c]; accd += v * d[c]; }
  as[i] = accs;
  ad[i] = accd;
}

// ------------------------------------------------------- edge pass 1: segment max
__global__ void edge_max_kernel(const int* __restrict__ ei, int E, int Etot,
                                const float* __restrict__ as,
                                const float* __restrict__ ad,
                                float* __restrict__ m) {
  int e = blockIdx.x * blockDim.x + threadIdx.x;
  if (e >= Etot) return;
  int src = (e < E) ? ei[e]     : (e - E);   // self-loops appended
  int dst = (e < E) ? ei[E + e] : (e - E);
#pragma unroll
  for (int h = 0; h < 4; ++h) {
    float v = as[src * 4 + h] + ad[dst * 4 + h];
    v = (v > 0.0f) ? v : NEG_SLOPE * v;
    atomicMaxFloat(&m[dst * 4 + h], v);
  }
}

// ------------------------------------- edge pass 2: exp(e - max), denom accumulation
__global__ void edge_exp_kernel(const int* __restrict__ ei, int E, int Etot,
                                const float* __restrict__ as,
                                const float* __restrict__ ad,
                                const float* __restrict__ m,
                                float* __restrict__ den,
                                float* __restrict__ ee) {
  int e = blockIdx.x * blockDim.x + threadIdx.x;
  if (e >= Etot) return;
  int src = (e < E) ? ei[e]     : (e - E);
  int dst = (e < E) ? ei[E + e] : (e - E);
#pragma unroll
  for (int h = 0; h < 4; ++h) {
    float v = as[src * 4 + h] + ad[dst * 4 + h];
    v = (v > 0.0f) ? v : NEG_SLOPE * v;
    float ex = expf(v - m[dst * 4 + h]);
    ee[e * 4 + h] = ex;
    atomicAdd(&den[dst * 4 + h], ex);
  }
}

// ------------------------- edge pass 3: weighted message scatter (block per edge)
// blockDim = HC (256 for layer1, 64 for layer2); thread t = channel index
__global__ void edge_aggr_kernel(const int* __restrict__ ei, int E,
                                 const float* __restrict__ hsrc,
                                 const float* __restrict__ ee,
                                 const float* __restrict__ den,
                                 float* __restrict__ out, int HC, int C) {
  int e = blockIdx.x;
  int t = threadIdx.x;
  int src = (e < E) ? ei[e]     : (e - E);
  int dst = (e < E) ? ei[E + e] : (e - E);
  int head = t / C;
  float alpha = ee[e * 4 + head] / den[dst * 4 + head];
  atomicAdd(&out[(long)dst * HC + t], alpha * hsrc[(long)src * HC + t]);
}

// -------------------------------------------------------------- bias + ReLU (in place)
__global__ void bias_relu_kernel(float* __restrict__ p, const float* __restrict__ b,
                                 long n, int F) {
  long i = (long)blockIdx.x * blockDim.x + threadIdx.x;
  if (i >= n) return;
  float v = p[i] + b[i % F];
  p[i] = (v > 0.0f) ? v : 0.0f;
}

// -------------------------------------------------------------- mean pool (sums+counts)
__global__ void pool_kernel(const float* __restrict__ h, const int* __restrict__ batch,
                            float* __restrict__ sums, float* __restrict__ counts, int N) {
  int n = blockIdx.x;
  int t = threadIdx.x;            // 64 threads = feature channel
  int g = batch[n];
  atomicAdd(&sums[g * 64 + t], h[(long)n * 64 + t]);
  if (t == 0) atomicAdd(&counts[g], 1.0f);
}

// -------------------------------------------------------------- final FC: [64,64]@[64,32]
__global__ void fc_kernel(const float* __restrict__ sums, const float* __restrict__ counts,
                          const float* __restrict__ w, const float* __restrict__ bias,
                          float* __restrict__ out) {
  int g = blockIdx.x;             // 64 graphs
  int o = threadIdx.x;            // 32 outputs
  float cnt = counts[g];
  cnt = (cnt > 1.0f) ? cnt : 1.0f;
  float inv = 1.0f / cnt;
  float acc = bias[o];
  for (int c = 0; c < 64; ++c)
    acc += (sums[g * 64 + c] * inv) * w[c * 32 + o];
  out[g * 32 + o] = acc;
}

// ============================================================== host driver
extern "C" void kernel_launch(void* const* d_in, const int* in_sizes, int n_in,
                              void* d_out, int out_size, void* d_ws, size_t ws_size,
                              hipStream_t stream) {
  const float* x      = (const float*)d_in[0];
  const int*   ei     = (const int*)  d_in[1];
  const int*   batch  = (const int*)  d_in[2];
  const float* W1     = (const float*)d_in[3];
  const float* a_src1 = (const float*)d_in[4];
  const float* a_dst1 = (const float*)d_in[5];
  const float* b1     = (const float*)d_in[6];
  const float* W2     = (const float*)d_in[7];
  const float* a_src2 = (const float*)d_in[8];
  const float* a_dst2 = (const float*)d_in[9];
  const float* b2     = (const float*)d_in[10];
  const float* fcw    = (const float*)d_in[11];
  const float* fcb    = (const float*)d_in[12];
  float* out = (float*)d_out;

  const int N    = in_sizes[0] / 128;   // nodes
  const int E    = in_sizes[1] / 2;     // edges
  const int Etot = E + N;               // + self-loops
  const int F1 = 256, F2 = 64;

  // workspace layout (floats)
  float* ws = (float*)d_ws;
  long off = 0;
  float* h1   = ws + off; off += (long)N * F1;
  float* out1 = ws + off; off += (long)N * F1;
  float* h2   = ws + off; off += (long)N * F2;
  float* out2 = ws + off; off += (long)N * F2;
  float* as_  = ws + off; off += (long)N * 4;
  float* ad_  = ws + off; off += (long)N * 4;
  float* mx   = ws + off; off += (long)N * 4;
  float* den  = ws + off; off += (long)N * 4;
  float* ee   = ws + off; off += (long)Etot * 4;
  float* sums = ws + off; off += 64 * 64;
  float* cnts = ws + off; off += 64;

  const int mt = (N + 15) / 16;
  dim3 wave(32);

  // ---------------- Layer 1 ----------------
  gemm_wmma_f32_kernel<<<dim3(mt, F1 / 64), wave, 0, stream>>>(x, W1, h1, N, 128, F1);
  alpha_kernel<<<(N * 4 + 255) / 256, 256, 0, stream>>>(h1, a_src1, a_dst1, as_, ad_, N, 64);
  fill_kernel<<<1024, 256, 0, stream>>>(mx, -INFINITY, (long)N * 4);
  fill_kernel<<<1024, 256, 0, stream>>>(den, 0.0f, (long)N * 4);
  fill_kernel<<<4096, 256, 0, stream>>>(out1, 0.0f, (long)N * F1);
  edge_max_kernel<<<(Etot + 255) / 256, 256, 0, stream>>>(ei, E, Etot, as_, ad_, mx);
  edge_exp_kernel<<<(Etot + 255) / 256, 256, 0, stream>>>(ei, E, Etot, as_, ad_, mx, den, ee);
  edge_aggr_kernel<<<Etot, F1, 0, stream>>>(ei, E, h1, ee, den, out1, F1, 64);
  bias_relu_kernel<<<(int)(((long)N * F1 + 255) / 256), 256, 0, stream>>>(out1, b1, (long)N * F1, F1);

  // ---------------- Layer 2 ----------------
  gemm_wmma_f32_kernel<<<dim3(mt, F2 / 64), wave, 0, stream>>>(out1, W2, h2, N, F1, F2);
  alpha_kernel<<<(N * 4 + 255) / 256, 256, 0, stream>>>(h2, a_src2, a_dst2, as_, ad_, N, 16);
  fill_kernel<<<1024, 256, 0, stream>>>(mx, -INFINITY, (long)N * 4);
  fill_kernel<<<1024, 256, 0, stream>>>(den, 0.0f, (long)N * 4);
  fill_kernel<<<2048, 256, 0, stream>>>(out2, 0.0f, (long)N * F2);
  edge_max_kernel<<<(Etot + 255) / 256, 256, 0, stream>>>(ei, E, Etot, as_, ad_, mx);
  edge_exp_kernel<<<(Etot + 255) / 256, 256, 0, stream>>>(ei, E, Etot, as_, ad_, mx, den, ee);
  edge_aggr_kernel<<<Etot, F2, 0, stream>>>(ei, E, h2, ee, den, out2, F2, 16);
  bias_relu_kernel<<<(int)(((long)N * F2 + 255) / 256), 256, 0, stream>>>(out2, b2, (long)N * F2, F2);

  // ---------------- Pool + FC ----------------
  fill_kernel<<<16, 256, 0, stream>>>(sums, 0.0f, (long)(64 * 64 + 64)); // sums + cnts contiguous
  pool_kernel<<<N, 64, 0, stream>>>(out2, batch, sums, cnts, N);
  fc_kernel<<<64, 32, 0, stream>>>(sums, cnts, fcw, fcb, out);
}